// OutputProcessPerObjectType_79190607004038
// MI455X (gfx1250) — compile-verified
//
#include <hip/hip_runtime.h>

typedef __attribute__((ext_vector_type(2))) float v2f;
typedef __attribute__((ext_vector_type(4))) float v4f;
typedef __attribute__((ext_vector_type(8))) float v8f;

#define NFRAMES     196
#define BATCH       128
#define LATENT      1024
#define NEXPERT     16
#define NFEATS      6
#define INPUT_FEATS 768

#define BM 32      // block tile over frames (M)
#define BN 64      // block tile over features (N)
#define BK 32      // K chunk staged in LDS
#define AS_STRIDE 36   // 32 + 4 pad: conflict-free column reads, 16B-aligned rows
#define BS_STRIDE 68   // 64 + 4 pad, 16B-aligned rows

// Async global->LDS copy (CDNA5): data lands in LDS without a VGPR round-trip.
// VDST = 32-bit LDS byte address (low 32 bits of the generic pointer per the
// flat->LDS aperture mapping), VADDR = 64-bit global address, no SADDR.
__device__ __forceinline__ void async_copy_b128(const float* g, unsigned lds_addr) {
    asm volatile("global_load_async_to_lds_b128 %0, %1, off"
                 :: "v"(lds_addr), "v"((unsigned long long)(uintptr_t)g)
                 : "memory");
}

__global__ __launch_bounds__(256)
void moe_head_wmma_f32(const float* __restrict__ out_lat,     // [196,128,1024]
                       const int*   __restrict__ obj_type,    // [128]
                       const float* __restrict__ W,           // [16,1024,768]
                       const float* __restrict__ bias_vec,    // [16,768]
                       const int*   __restrict__ joint_counts,// [16]
                       float*       __restrict__ y)           // [128,768,196]
{
    __shared__ float As[2][BM * AS_STRIDE];
    __shared__ float Bs[2][BK * BS_STRIDE];

    const int b  = blockIdx.x;   // sample
    const int bm = blockIdx.y;   // M tile (0..6)
    const int bn = blockIdx.z;   // N tile (0..11)

    const int tid  = threadIdx.x;
    const int lane = tid & 31;
    const int wid  = tid >> 5;   // 0..7
    const int wm   = wid >> 2;   // 0..1 : wave's 16-row band inside BM
    const int wn   = wid & 3;    // 0..3 : wave's 16-col band inside BN

    const int e = obj_type[b];
    const float* Wb = W + (size_t)e * LATENT * INPUT_FEATS + bn * BN;
    const float* Ab = out_lat + (size_t)b * LATENT;   // + t * (BATCH*LATENT)

    // Cooperative async-copy coordinates: each thread moves 3 x 16B per chunk.
    const int a_row  = tid >> 3;          // 0..31 (frame row inside tile)
    const int a_col4 = (tid & 7) << 2;    // 0,4,...,28 (K offset)
    const int t_row  = bm * BM + a_row;   // global frame index for A copy
    const int t_safe = (t_row < NFRAMES) ? t_row : (NFRAMES - 1); // clamp: no OOB,
                                          // garbage rows feed only t>=196 outputs (never stored)
    const int b_row  = tid >> 4;          // 0..15 (K row inside tile; +16 second half)
    const int b_col4 = (tid & 15) << 2;   // 0..60 (feature offset)

    const size_t a_gstride = (size_t)BATCH * LATENT;  // frame stride of output[]

    // Per-thread LDS destinations for the async copies, both buffers.
    unsigned lds_a[2], lds_b0[2], lds_b1[2];
    #pragma unroll
    for (int p = 0; p < 2; ++p) {
        lds_a[p]  = (unsigned)(uintptr_t)&As[p][a_row * AS_STRIDE + a_col4];
        lds_b0[p] = (unsigned)(uintptr_t)&Bs[p][b_row * BS_STRIDE + b_col4];
        lds_b1[p] = (unsigned)(uintptr_t)&Bs[p][(b_row + 16) * BS_STRIDE + b_col4];
    }

    v8f acc = {};

    // Per-lane WMMA fragment coordinates (ISA 16x4 f32 layout):
    // lane L: M = L%16 ; VGPR v holds K = 2*(L/16) + v. B mirrored in N.
    const int frag_m = (lane & 15) + wm * 16;
    const int frag_c = (lane & 15) + wn * 16;
    const int khalf  = (lane >> 4) << 1;  // 0 or 2

    // ---- prologue: issue chunk 0 into buffer 0 ----
    async_copy_b128(Ab + (size_t)t_safe * a_gstride + a_col4,                 lds_a[0]);
    async_copy_b128(Wb + (size_t)b_row * INPUT_FEATS + b_col4,                lds_b0[0]);
    async_copy_b128(Wb + (size_t)(b_row + 16) * INPUT_FEATS + b_col4,         lds_b1[0]);

    for (int k0 = 0; k0 < LATENT; k0 += BK) {
        const int cur = (k0 >> 5) & 1;

        if (k0 + BK < LATENT) {
            // issue next chunk into the other buffer (freed by previous
            // iteration's trailing barrier), then wait for current chunk:
            // the 3 most recent async ops are the next chunk's.
            const int nk  = k0 + BK;
            const int nxt = cur ^ 1;
            async_copy_b128(Ab + (size_t)t_safe * a_gstride + nk + a_col4,            lds_a[nxt]);
            async_copy_b128(Wb + (size_t)(nk + b_row) * INPUT_FEATS + b_col4,         lds_b0[nxt]);
            async_copy_b128(Wb + (size_t)(nk + b_row + 16) * INPUT_FEATS + b_col4,    lds_b1[nxt]);
            if (k0 + 2 * BK < LATENT)  // L2 warm-up two chunks ahead for the reused W
                __builtin_prefetch(Wb + (size_t)(k0 + 2 * BK + b_row) * INPUT_FEATS + b_col4, 0, 1);
            asm volatile("s_wait_asynccnt 0x3" ::: "memory");
        } else {
            asm volatile("s_wait_asynccnt 0x0" ::: "memory");
        }
        __syncthreads();   // current chunk visible to all waves

        // ---- 8 x v_wmma_f32_16x16x4_f32 over the 32-deep chunk ----
        #pragma unroll
        for (int kk = 0; kk < BK; kk += 4) {
            v2f af, bf;
            af[0] = As[cur][frag_m * AS_STRIDE + kk + khalf];
            af[1] = As[cur][frag_m * AS_STRIDE + kk + khalf + 1];
            bf[0] = Bs[cur][(kk + khalf)     * BS_STRIDE + frag_c];
            bf[1] = Bs[cur][(kk + khalf + 1) * BS_STRIDE + frag_c];
            acc = __builtin_amdgcn_wmma_f32_16x16x4_f32(
                /*neg_a=*/false, af, /*neg_b=*/false, bf,
                /*c_mod=*/(short)0, acc, /*reuse_a=*/false, /*reuse_b=*/false);
        }
        __syncthreads();   // all waves done reading 'cur' before it is refilled
    }

    // ---- epilogue: bias + column mask, store transposed (t contiguous) ----
    const int f     = bn * BN + wn * 16 + (lane & 15);
    const int fc    = joint_counts[e] * NFEATS;
    const float bv  = bias_vec[e * INPUT_FEATS + f];
    const float msk = (f < fc) ? 1.0f : 0.0f;

    // acc[v] holds M = v + 8*(lane/16)  ->  t = t_base + v(mod 8) contiguous
    const int t_base = bm * BM + wm * 16 + ((lane >> 4) << 3);
    float* yp = y + (size_t)(b * INPUT_FEATS + f) * NFRAMES + t_base;

    v4f lo, hi;
    lo[0] = (acc[0] + bv) * msk;  lo[1] = (acc[1] + bv) * msk;
    lo[2] = (acc[2] + bv) * msk;  lo[3] = (acc[3] + bv) * msk;
    hi[0] = (acc[4] + bv) * msk;  hi[1] = (acc[5] + bv) * msk;
    hi[2] = (acc[6] + bv) * msk;  hi[3] = (acc[7] + bv) * msk;

    if (t_base + 7 < NFRAMES) {
        *(v4f*)(yp)     = lo;
        *(v4f*)(yp + 4) = hi;
    } else {
        #pragma unroll
        for (int v = 0; v < 4; ++v)
            if (t_base + v < NFRAMES) yp[v] = lo[v];
        #pragma unroll
        for (int v = 4; v < 8; ++v)
            if (t_base + v < NFRAMES) yp[v] = hi[v - 4];
    }
}

extern "C" void kernel_launch(void* const* d_in, const int* in_sizes, int n_in,
                              void* d_out, int out_size, void* d_ws, size_t ws_size,
                              hipStream_t stream) {
    const float* out_lat      = (const float*)d_in[0]; // [196,128,1024]
    const int*   object_types = (const int*)  d_in[1]; // [128]
    const float* W            = (const float*)d_in[2]; // [16,1024,768]
    const float* bias_vec     = (const float*)d_in[3]; // [16,768]
    const int*   joint_counts = (const int*)  d_in[4]; // [16]
    float*       y            = (float*)d_out;         // [128,768,196]

    dim3 grid(BATCH, (NFRAMES + BM - 1) / BM, INPUT_FEATS / BN); // 128 x 7 x 12
    moe_head_wmma_f32<<<grid, 256, 0, stream>>>(
        out_lat, object_types, W, bias_vec, joint_counts, y);
}